// HungarianMatcher_63608465654575
// MI455X (gfx1250) — compile-verified
//
#include <hip/hip_runtime.h>
#include <hip/hip_bf16.h>

// Hungarian-matcher cost matrix for MI455X (gfx1250, wave32).
//
// total[b,q,n] = -softmax(logits[b,q])[label[b,n]]
//              + 5*dist(q,n)        (inf -> 1e6 when dist > 2)
//              + 2*(1 - IoU3D(q,n))
//
// Store-bandwidth bound (256 MB output, ~11us floor at 23.3 TB/s). Class cost
// done as a one-hot GEMM with V_WMMA_F32_16X16X4_F32 (K padded 21->24, 6 WMMA
// steps per 16x16 tile). Epilogue uses raw v_sqrt_f32 / v_rcp_f32 to avoid the
// IEEE-correct expansion sequences (~1 ulp, fine for a matching cost).

typedef __attribute__((ext_vector_type(2))) float v2f;
typedef __attribute__((ext_vector_type(8))) float v8f;

constexpr int   kB  = 64;
constexpr int   kQ  = 4096;
constexpr int   kN  = 256;
constexpr int   kC  = 21;   // NUM_CLASSES + 1 (background last)
constexpr int   kFG = 20;
constexpr float kCostCenter = 5.0f;
constexpr float kCostIou    = 2.0f;
constexpr float kCenterTh   = 2.0f;
constexpr float kInfReplace = 1000000.0f;

__global__ __launch_bounds__(256)
void matcher_cost_kernel(const float* __restrict__ pred_logits,   // [B,Q,21]
                         const float* __restrict__ pred_boxes,    // [B,Q,6]
                         const int*   __restrict__ target_labels, // [B,N]
                         const float* __restrict__ target_boxes,  // [B,N,6]
                         float*       __restrict__ out)           // [B,Q,N]
{
  const int lane = threadIdx.x & 31;
  const int wave = threadIdx.x >> 5;
  const int qt   = blockIdx.x * 8 + wave;   // q-tile index, 0..255
  const int b    = blockIdx.y;
  const int q0   = qt * 16;
  const int m    = lane & 15;               // matrix row/col this lane serves
  const int half = lane >> 4;               // 0: lanes 0-15, 1: lanes 16-31

  // ---- softmax for query row (q0+m); both lane-halves redundantly hold rows 0..15,
  //      which is exactly the replication the 16x4 f32 A-matrix layout wants. ----
  const float* lg = pred_logits + ((size_t)b * kQ + (q0 + m)) * kC;
  float l[kC];
#pragma unroll
  for (int c = 0; c < kC; ++c) l[c] = lg[c];
  float mx = l[0];
#pragma unroll
  for (int c = 1; c < kC; ++c) mx = fmaxf(mx, l[c]);
  float sum = 0.0f;
#pragma unroll
  for (int c = 0; c < kC; ++c) { l[c] = __expf(l[c] - mx); sum += l[c]; }
  const float inv = __builtin_amdgcn_rcpf(sum);     // sum >= 1, v_rcp_f32 is ~1 ulp
  float p[24];
#pragma unroll
  for (int c = 0; c < kFG; ++c) p[c] = l[c] * inv;  // drop background class
  p[20] = p[21] = p[22] = p[23] = 0.0f;             // pad K: 21 -> 24

  // A fragments for the 6 WMMA K-steps.
  // Documented 16x4 f32 A layout: VGPR0 = K{0|2}, VGPR1 = K{1|3} per lane-half.
  v2f A[6];
#pragma unroll
  for (int k = 0; k < 6; ++k) {
    A[k].x = half ? p[4 * k + 2] : p[4 * k + 0];
    A[k].y = half ? p[4 * k + 3] : p[4 * k + 1];
  }

  // ---- pred-box geometry for the 8 output rows this lane-half covers.
  //      D-matrix layout: VGPR r -> M = r + 8*half. ----
  const float* pb = pred_boxes + ((size_t)b * kQ + q0 + 8 * half) * 6;
  float pcx[8], pcy[8], pcz[8], psx[8], psy[8], psz[8];
#pragma unroll
  for (int r = 0; r < 8; ++r) {
    pcx[r] = pb[r * 6 + 0]; pcy[r] = pb[r * 6 + 1]; pcz[r] = pb[r * 6 + 2];
    psx[r] = pb[r * 6 + 3]; psy[r] = pb[r * 6 + 4]; psz[r] = pb[r * 6 + 5];
  }

  const int*   lb = target_labels + b * kN;
  const float* tb = target_boxes + (size_t)b * kN * 6;
  float*       ob = out + ((size_t)b * kQ + q0 + 8 * half) * kN;

  for (int nt = 0; nt < 16; ++nt) {
    const int n   = nt * 16 + m;            // this lane's target column
    const int lab = lb[n];

    const float* tr = tb + n * 6;
    const float tcx = tr[0], tcy = tr[1], tcz = tr[2];
    const float tsx = tr[3], tsy = tr[4], tsz = tr[5];
    const float tminx = tcx - 0.5f * tsx, tmaxx = tcx + 0.5f * tsx;
    const float tminy = tcy - 0.5f * tsy, tmaxy = tcy + 0.5f * tsy;
    const float tminz = tcz - 0.5f * tsz, tmaxz = tcz + 0.5f * tsz;
    const float tvol  = tsx * tsy * tsz;

    // ---- class cost tile via one-hot GEMM: D = P(16x24) x OneHot(24x16) ----
    v8f acc = {};
#pragma unroll
    for (int k = 0; k < 6; ++k) {
      v2f Bv;  // one-hot column for this lane's target; same half/K striping as A
      Bv.x = (lab == (4 * k + 2 * half))     ? 1.0f : 0.0f;
      Bv.y = (lab == (4 * k + 2 * half + 1)) ? 1.0f : 0.0f;
      acc = __builtin_amdgcn_wmma_f32_16x16x4_f32(
          /*neg_a=*/false, A[k], /*neg_b=*/false, Bv,
          /*c_mod=*/(short)0, acc, /*reuse_a=*/false, /*reuse_b=*/false);
    }

    // ---- epilogue: distance + IoU + combine, coalesced stores ----
#pragma unroll
    for (int r = 0; r < 8; ++r) {
      const float dx = pcx[r] - tcx, dy = pcy[r] - tcy, dz = pcz[r] - tcz;
      const float d2   = dx * dx + dy * dy + dz * dz;
      const float dist = __builtin_amdgcn_sqrtf(d2);   // raw v_sqrt_f32

      float ix = fminf(pcx[r] + 0.5f * psx[r], tmaxx) -
                 fmaxf(pcx[r] - 0.5f * psx[r], tminx);
      float iy = fminf(pcy[r] + 0.5f * psy[r], tmaxy) -
                 fmaxf(pcy[r] - 0.5f * psy[r], tminy);
      float iz = fminf(pcz[r] + 0.5f * psz[r], tmaxz) -
                 fmaxf(pcz[r] - 0.5f * psz[r], tminz);
      ix = fmaxf(ix, 0.0f); iy = fmaxf(iy, 0.0f); iz = fmaxf(iz, 0.0f);

      const float inter = ix * iy * iz;
      const float pvol  = psx[r] * psy[r] * psz[r];
      const float uni   = pvol + tvol - inter;
      // sizes >= 0.5 so uni >= ~0.25 always; raw v_rcp_f32 (~1 ulp)
      const float iou   = (uni > 0.0f) ? inter * __builtin_amdgcn_rcpf(uni) : 0.0f;

      float total = -acc[r] + kCostCenter * dist + kCostIou * (1.0f - iou);
      total = (dist <= kCenterTh) ? total : kInfReplace;

      ob[(size_t)r * kN + n] = total;
    }
  }
}

extern "C" void kernel_launch(void* const* d_in, const int* in_sizes, int n_in,
                              void* d_out, int out_size, void* d_ws, size_t ws_size,
                              hipStream_t stream) {
  const float* pred_logits   = (const float*)d_in[0];
  const float* pred_boxes    = (const float*)d_in[1];
  const int*   target_labels = (const int*)d_in[2];
  const float* target_boxes  = (const float*)d_in[3];
  float*       out           = (float*)d_out;

  dim3 grid(kQ / 16 / 8, kB);   // 8 q-tiles (waves) per 256-thread block
  matcher_cost_kernel<<<grid, 256, 0, stream>>>(
      pred_logits, pred_boxes, target_labels, target_boxes, out);
}